// DenselyCnnAttLayer_21715354648961
// MI455X (gfx1250) — compile-verified
//
#include <hip/hip_runtime.h>
#include <math.h>

// Problem constants (reference: B=64, S=512, L=6, D=512)
#define B_ 64
#define S_ 512
#define L_ 6
#define D_ 512
#define NPOS (B_ * S_)          // 32768 positions

#define WPB 2                   // waves per block (wave32)
#define TPB (WPB * 32)          // 64 threads
#define NBLOCKS 4096            // 8192 waves -> 4 positions per wave

typedef float v2f __attribute__((ext_vector_type(2)));
typedef float v4f __attribute__((ext_vector_type(4)));
typedef float v8f __attribute__((ext_vector_type(8)));

// s_wait_asynccnt <imm> : wait until ASYNCcnt <= n (async loads complete in order)
#define WAIT_ASYNC(n) asm volatile("s_wait_asynccnt " #n ::: "memory")

// Get 32-bit LDS byte offset of a generic pointer into shared memory.
__device__ __forceinline__ unsigned lds_off_u32(void* p) {
  return (unsigned)(unsigned long long)(__attribute__((address_space(3))) void*)p;
}

// Asynchronously copy one position's 6x512 f32 tile (12 KB) from global to LDS.
// 24 x global_load_async_to_lds_b128: each lane moves 16B per instruction.
__device__ __forceinline__ void async_fetch_pos(const float* const xs[L_], int pos,
                                                float* ldsbuf, int lane) {
  const unsigned base = lds_off_u32(ldsbuf) + (unsigned)lane * 16u;
#pragma unroll
  for (int l = 0; l < L_; ++l) {
    const float* g = xs[l] + (size_t)pos * D_ + lane * 4;
#pragma unroll
    for (int c = 0; c < 4; ++c) {
      asm volatile("global_load_async_to_lds_b128 %0, %1, off"
                   :: "v"(base + (unsigned)((l * D_ + c * 128) * 4)),
                      "v"(g + c * 128)
                   : "memory");
    }
  }
}

__device__ __forceinline__ void compute_pos(const float* buf, const float* __restrict__ Ws,
                                            float* __restrict__ out, int pos, int lane) {
  const v4f* bv = (const v4f*)buf;  // 6 rows x 128 v4f

  // ---- Phase A: per-layer feature sums (wave32 butterfly reduction) ----
  float sums[L_];
#pragma unroll
  for (int l = 0; l < L_; ++l) {
    float s = 0.f;
#pragma unroll
    for (int c = 0; c < 4; ++c) {
      v4f v = bv[l * 128 + c * 32 + lane];
      s += (v.x + v.y) + (v.z + v.w);
    }
#pragma unroll
    for (int off = 16; off > 0; off >>= 1) s += __shfl_xor(s, off, 32);
    sums[l] = s;  // every lane holds the full sum after butterfly
  }

  // ---- Phase B: logits = sums(1x6) . Ws[sp](6x6) on the matrix unit ----
  // V_WMMA_F32_16X16X4_F32, two K-steps (K=0..3, K=4..5 with K=6,7 zero-padded).
  // A (16x4 f32): lanes 0-15 carry K=0 (v0) / K=1 (v1); lanes 16-31 carry K=2/K=3.
  // All 16 M-rows replicated with the sums vector -> every D row equals logits.
  // B (4x16 f32): lane n (0..15) = column; v0 = K row {0|2}, v1 = K row {1|3}.
  // D (16x16 f32): lane m (0..15), VGPR0 = row M=0 -> logit_m sits in C[0] of lane m.
  const int sp = pos & (S_ - 1);  // S_ = 512 is a power of two
  const int n  = lane & 15;
  const int hi = lane >> 4;                 // 0: lanes 0-15, 1: lanes 16-31
  const bool nvalid = (n < L_);
  // Clamped, always-in-bounds column pointer (max index sp*36 + 35 < S*36).
  const float* wcol = Ws + sp * (L_ * L_) + (nvalid ? n : 0);

  float b10 = wcol[(hi ? 2 : 0) * L_];      // K = 0 | 2
  float b11 = wcol[(hi ? 3 : 1) * L_];      // K = 1 | 3
  float b20 = wcol[4 * L_];                 // K = 4 (upper half-wave zeroed below)
  float b21 = wcol[5 * L_];                 // K = 5
  if (!nvalid) { b10 = 0.f; b11 = 0.f; }
  if (!nvalid || hi) { b20 = 0.f; b21 = 0.f; }

  v2f A1; A1.x = hi ? sums[2] : sums[0]; A1.y = hi ? sums[3] : sums[1];
  v2f A2; A2.x = hi ? 0.f : sums[4];     A2.y = hi ? 0.f : sums[5];
  v2f B1; B1.x = b10; B1.y = b11;
  v2f B2; B2.x = b20; B2.y = b21;

  v8f Cc = {};
  Cc = __builtin_amdgcn_wmma_f32_16x16x4_f32(false, A1, false, B1, (short)0, Cc,
                                             false, false);
  Cc = __builtin_amdgcn_wmma_f32_16x16x4_f32(false, A2, false, B2, (short)0, Cc,
                                             false, false);

  // ---- Softmax over the 6 logits (lanes 0..5 hold logit_m in Cc[0]) ----
  float logit = (lane < L_) ? Cc[0] : -INFINITY;
  float mx = logit;
#pragma unroll
  for (int off = 4; off > 0; off >>= 1) mx = fmaxf(mx, __shfl_xor(mx, off, 8));
  float e = (lane < L_) ? __expf(logit - mx) : 0.f;
  float den = e;
#pragma unroll
  for (int off = 4; off > 0; off >>= 1) den += __shfl_xor(den, off, 8);
  const float aw = e / den;  // valid on lanes 0..5

  float a[L_];
#pragma unroll
  for (int m = 0; m < L_; ++m) a[m] = __shfl(aw, m, 32);

  // ---- Phase C: out[d] = sum_l a_l * x_l[d]; NT store (write-once stream) ----
  v4f* o = (v4f*)(out + (size_t)pos * D_);
#pragma unroll
  for (int c = 0; c < 4; ++c) {
    float rx = 0.f, ry = 0.f, rz = 0.f, rw = 0.f;
#pragma unroll
    for (int l = 0; l < L_; ++l) {
      v4f v = bv[l * 128 + c * 32 + lane];
      rx = fmaf(a[l], v.x, rx);
      ry = fmaf(a[l], v.y, ry);
      rz = fmaf(a[l], v.z, rz);
      rw = fmaf(a[l], v.w, rw);
    }
    v4f r; r.x = rx; r.y = ry; r.z = rz; r.w = rw;
    __builtin_nontemporal_store(r, o + c * 32 + lane);
  }
}

__global__ __launch_bounds__(TPB) void densely_att_kernel(
    const float* __restrict__ x0, const float* __restrict__ x1,
    const float* __restrict__ x2, const float* __restrict__ x3,
    const float* __restrict__ x4, const float* __restrict__ x5,
    const float* __restrict__ Ws, float* __restrict__ out) {
  // 2 waves x 2 buffers x (6*512) floats = 48 KB static LDS
  __shared__ float lds[WPB][2][L_ * D_];

  const int lane = threadIdx.x & 31;
  const int wib  = threadIdx.x >> 5;
  const int waveId = blockIdx.x * WPB + wib;
  const int stride = gridDim.x * WPB;
  const float* xs[L_] = {x0, x1, x2, x3, x4, x5};

  int pos = waveId;
  if (pos >= NPOS) return;

  int cur = 0;
  async_fetch_pos(xs, pos, lds[wib][0], lane);  // prime the pipeline (24 async ops)

  while (true) {
    const int nxt = pos + stride;
    if (nxt < NPOS) {
      async_fetch_pos(xs, nxt, lds[wib][cur ^ 1], lane);  // 24 more in flight
      WAIT_ASYNC(24);  // current buffer's 24 loads done; prefetch still flying
    } else {
      WAIT_ASYNC(0);
    }
    compute_pos(lds[wib][cur], Ws, out, pos, lane);
    if (nxt >= NPOS) break;
    cur ^= 1;
    pos = nxt;
  }
}

extern "C" void kernel_launch(void* const* d_in, const int* in_sizes, int n_in,
                              void* d_out, int out_size, void* d_ws, size_t ws_size,
                              hipStream_t stream) {
  (void)in_sizes; (void)n_in; (void)out_size; (void)d_ws; (void)ws_size;
  const float* x0 = (const float*)d_in[0];
  const float* x1 = (const float*)d_in[1];
  const float* x2 = (const float*)d_in[2];
  const float* x3 = (const float*)d_in[3];
  const float* x4 = (const float*)d_in[4];
  const float* x5 = (const float*)d_in[5];
  const float* Ws = (const float*)d_in[6];
  float* out = (float*)d_out;

  densely_att_kernel<<<NBLOCKS, TPB, 0, stream>>>(x0, x1, x2, x3, x4, x5, Ws, out);
}